// FarthestPointSampler_11647951307128
// MI455X (gfx1250) — compile-verified
//
#include <hip/hip_runtime.h>
#include <stdint.h>

// Farthest point sampling for MI455X (gfx1250, wave32).
// One 1024-thread block (32 waves) per batch; dist array in d_ws (L2-resident,
// pos 96MB + dist 32MB = 128MB < 192MB L2). Per round: stream pos+dist from L2,
// min-update, block argmax via wave32 shfl_xor butterfly + LDS combine.

#define NP_     1024        // samples per batch
#define THREADS 1024        // 32 wave32 waves per block
#define KPT_FIXED 128       // 131072 / 1024 (register-fallback kernel only)

// distances are >= 0, so float bits compare like the floats themselves.
// pack = (dist_bits << 32) | ~index  ->  u64 max == (max dist, tie -> min index),
// exactly jnp.argmax first-occurrence semantics.
__device__ __forceinline__ uint64_t fps_pack(float d, unsigned i) {
    return (((uint64_t)__float_as_uint(d)) << 32) | (uint64_t)(~i);
}

__device__ __forceinline__ uint64_t fps_wave_max(uint64_t v) {
#pragma unroll
    for (int off = 16; off > 0; off >>= 1) {
        uint64_t o = (uint64_t)__shfl_xor((unsigned long long)v, off, 32);
        v = (o > v) ? o : v;
    }
    return v;
}

__global__ __launch_bounds__(THREADS) void fps_ws_kernel(
    const float* __restrict__ pos, const int* __restrict__ start_ptr,
    int* __restrict__ out, float* __restrict__ dist, int N)
{
#pragma clang fp contract(off)   // match XLA: no FMA contraction in (p-q)^2 sum
    __shared__ uint64_t su[THREADS / 32];
    __shared__ uint64_t sres;

    const int b    = blockIdx.x;
    const int tid  = threadIdx.x;
    const int lane = tid & 31;
    const int wv   = tid >> 5;

    const float* posb  = pos  + (size_t)b * 3u * (size_t)N;
    float*       distb = dist + (size_t)b * (size_t)N;

    // dist0 = +inf (reference). Each thread only ever touches its own strided
    // slots, so no barrier is needed between this init and the main loop.
    for (int i = tid; i < N; i += THREADS) distb[i] = INFINITY;

    // cur is uniform across the block; keep it in an SGPR so the q-point
    // address math stays on the SALU and q loads become scalar (KMcnt) loads.
    int cur = __builtin_amdgcn_readfirstlane(start_ptr[0]);

    for (int t = 0; t < NP_; ++t) {
        if (tid == 0) out[b * NP_ + t] = cur;
        if (t == NP_ - 1) break;

        const float* q = posb + 3u * (size_t)(unsigned)cur;
        const float qx = q[0];
        const float qy = q[1];
        const float qz = q[2];

        float    bestd = -INFINITY;
        unsigned besti = 0;
#pragma unroll 8
        for (int i = tid; i < N; i += THREADS) {
            const float* p = posb + 3u * (size_t)(unsigned)i;
            float dx = p[0] - qx;
            float dy = p[1] - qy;
            float dz = p[2] - qz;
            float d  = dx * dx + dy * dy + dz * dz;  // left-assoc, no FMA
            float nd = fminf(distb[i], d);
            distb[i] = nd;
            if (nd > bestd) { bestd = nd; besti = (unsigned)i; }  // strict >: first max
        }

        // wave32 butterfly reduce, then 32 wave leaders -> LDS -> wave 0 reduce
        uint64_t bp = fps_wave_max(fps_pack(bestd, besti));
        if (lane == 0) su[wv] = bp;
        __syncthreads();
        if (wv == 0) {
            uint64_t v = fps_wave_max(su[lane]);
            if (lane == 0) sres = v;
        }
        __syncthreads();
        cur = __builtin_amdgcn_readfirstlane((int)(~(unsigned)sres));
    }
}

// Fallback if d_ws is too small: keep the 128 per-thread distances in VGPRs
// (fully unrolled, compile-time indices). __launch_bounds__ forces the
// allocator to fit 32 waves/WGP (spilling instead of failing to launch).
__global__ __launch_bounds__(THREADS) void fps_reg_kernel(
    const float* __restrict__ pos, const int* __restrict__ start_ptr,
    int* __restrict__ out)
{
#pragma clang fp contract(off)
    __shared__ uint64_t su[THREADS / 32];
    __shared__ uint64_t sres;

    const int b = blockIdx.x, tid = threadIdx.x, lane = tid & 31, wv = tid >> 5;
    const float* posb = pos + (size_t)b * 3u * (size_t)(THREADS * KPT_FIXED);

    float dl[KPT_FIXED];
#pragma unroll
    for (int k = 0; k < KPT_FIXED; ++k) dl[k] = INFINITY;

    int cur = __builtin_amdgcn_readfirstlane(start_ptr[0]);
    for (int t = 0; t < NP_; ++t) {
        if (tid == 0) out[b * NP_ + t] = cur;
        if (t == NP_ - 1) break;

        const float* q = posb + 3u * (size_t)(unsigned)cur;
        const float qx = q[0];
        const float qy = q[1];
        const float qz = q[2];

        float    bestd = -INFINITY;
        unsigned besti = 0;
#pragma unroll
        for (int k = 0; k < KPT_FIXED; ++k) {
            unsigned i = (unsigned)tid + ((unsigned)k << 10);
            const float* p = posb + 3u * (size_t)i;
            float dx = p[0] - qx;
            float dy = p[1] - qy;
            float dz = p[2] - qz;
            float d  = dx * dx + dy * dy + dz * dz;
            float nd = fminf(dl[k], d);
            dl[k] = nd;
            if (nd > bestd) { bestd = nd; besti = i; }
        }

        uint64_t bp = fps_wave_max(fps_pack(bestd, besti));
        if (lane == 0) su[wv] = bp;
        __syncthreads();
        if (wv == 0) {
            uint64_t v = fps_wave_max(su[lane]);
            if (lane == 0) sres = v;
        }
        __syncthreads();
        cur = __builtin_amdgcn_readfirstlane((int)(~(unsigned)sres));
    }
}

extern "C" void kernel_launch(void* const* d_in, const int* in_sizes, int n_in,
                              void* d_out, int out_size, void* d_ws, size_t ws_size,
                              hipStream_t stream)
{
    const float* pos   = (const float*)d_in[0];   // (B, N, 3) float32
    const int*   start = (const int*)d_in[1];     // scalar start_idx
    int*         out   = (int*)d_out;             // (B, 1024) int32

    const int B = out_size / NP_;                 // 64
    const int N = in_sizes[0] / (3 * B);          // 131072
    const size_t need = (size_t)B * (size_t)N * sizeof(float);

    dim3 grid(B), block(THREADS);
    if (ws_size >= need) {
        fps_ws_kernel<<<grid, block, 0, stream>>>(pos, start, out, (float*)d_ws, N);
    } else {
        fps_reg_kernel<<<grid, block, 0, stream>>>(pos, start, out);
    }
}